// CoreAttention_36404142800928
// MI455X (gfx1250) — compile-verified
//
#include <hip/hip_runtime.h>
#include <cstdint>

typedef __attribute__((ext_vector_type(2))) float v2f;
typedef __attribute__((ext_vector_type(8))) float v8f;

#define HEADS 8
#define HDIM  16
#define TILE  16   // edges per wave
#define WAVES_PER_BLOCK 4

// ---------------------------------------------------------------------------
// Kernel 0: zero the accumulators (d_out = wV, d_ws = Z). Harness poisons them.
// ---------------------------------------------------------------------------
__global__ void ca_zero_kernel(float* __restrict__ out, int n_out,
                               float* __restrict__ z, int n_z) {
    int i = blockIdx.x * blockDim.x + threadIdx.x;
    if (i < n_out) out[i] = 0.0f;
    if (i < n_z)   z[i]   = 0.0f;
}

// ---------------------------------------------------------------------------
// Kernel 1: per-edge attention scores via WMMA + atomic scatter.
//
// One wave32 handles TILE=16 edges. For each head h:
//   A (16x4 f32)  row m = k[src[e_m], h*16+kk .. +3]
//   B (4x16 f32)  col n = q[dst[e_n], h*16+kk .. +3]
//   S = A x B accumulated over kk = 0,4,8,12  (V_WMMA_F32_16X16X4_F32)
// diag(S)[e] = dot(k[src[e]], q[dst[e]]) -> exp(clip(./4, -5, 5)) -> LDS.
//
// D layout (32-bit 16x16): VGPR r, lanes 0-15 -> (M=r, N=lane);
//                          lanes 16-31 -> (M=r+8, N=lane-16).
// Diagonal e<8  lives in VGPR e,   lane e.
// Diagonal e>=8 lives in VGPR e-8, lane e+16.
// => every diagonal-holding lane (0-7, 24-31) holds its element in
//    acc[lane & 7]: selected with a branch-free cndmask chain.
// ---------------------------------------------------------------------------
__global__ __launch_bounds__(WAVES_PER_BLOCK * 32)
void ca_edge_kernel(const float* __restrict__ q, const float* __restrict__ k,
                    const float* __restrict__ v,
                    const int64_t* __restrict__ ei,
                    float* __restrict__ wv, float* __restrict__ z,
                    int E) {
    __shared__ float ssc[WAVES_PER_BLOCK][HEADS * TILE];

    const int lane = threadIdx.x & 31;
    const int wid  = threadIdx.x >> 5;
    const int half = lane >> 4;     // 0: lanes 0-15, 1: lanes 16-31
    const int l16  = lane & 15;

    const int64_t* __restrict__ srcp = ei;      // edge_index[0][*]
    const int64_t* __restrict__ dstp = ei + E;  // edge_index[1][*]

    const int tile  = blockIdx.x * WAVES_PER_BLOCK + wid;
    const int ebase = tile * TILE;

    // Edge owned by this lane for the matrix loads (row/col index = l16).
    // Clamp out-of-range edges (selects, not branches: EXEC all-1 at WMMA).
    const int  eidx   = ebase + l16;
    const int  eclamp = (eidx < E) ? eidx : 0;
    const long sA = (long)srcp[eclamp];
    const long dB = (long)dstp[eclamp];

    const float* __restrict__ krow = k + sA * (HEADS * HDIM);
    const float* __restrict__ qrow = q + dB * (HEADS * HDIM);

    // Diagonal bookkeeping (invariant over heads).
    const int  r7     = lane & 7;
    const bool isdiag = (lane < 8) || (lane >= 24);
    const int  ediag  = (lane < 8) ? lane : (lane - 16);

    // ---- score phase: 8 heads x 4 WMMA steps --------------------------------
    for (int h = 0; h < HEADS; ++h) {
        v8f acc = {};
#pragma unroll
        for (int kk = 0; kk < HDIM; kk += 4) {
            const int off = h * HDIM + kk + half * 2;
            v2f a = *(const v2f*)(krow + off);
            v2f b = *(const v2f*)(qrow + off);
            // (neg_a, A, neg_b, B, c_mod, C, reuse_a, reuse_b)
            acc = __builtin_amdgcn_wmma_f32_16x16x4_f32(
                false, a, false, b, (short)0, acc, false, false);
        }
        // Branch-free select of this lane's diagonal element.
        float dv = acc[0];
#pragma unroll
        for (int r = 1; r < 8; ++r) dv = (r7 == r) ? acc[r] : dv;
        const float sc = __expf(fminf(fmaxf(dv * 0.25f, -5.0f), 5.0f));
        if (isdiag) ssc[wid][h * TILE + ediag] = sc;   // single predicated store
    }
    __syncthreads();

    // ---- Z scatter: 128 (edge, head) pairs, 4 per lane, branch-free ---------
#pragma unroll
    for (int i = 0; i < 4; ++i) {
        const int p  = i * 32 + lane;    // 0..127
        const int e  = p >> 3;           // edge in tile
        const int h  = p & 7;            // head
        const int eg = ebase + e;
        const bool ok = (eg < E);
        const long t  = (long)dstp[ok ? eg : 0];
        const float sc = ssc[wid][h * TILE + e];
        if (ok) atomicAdd(&z[t * HEADS + h], sc);
    }

    // ---- wV scatter: one edge per iteration, 4 floats per lane --------------
    for (int e = 0; e < TILE; ++e) {
        const int eg = ebase + e;
        if (eg >= E) break;              // wave-uniform: scalar branch
        const long s = (long)srcp[eg];
        const long t = (long)dstp[eg];
        const float4 vv = *(const float4*)(v + s * (HEADS * HDIM) + lane * 4);
        const float  sc = ssc[wid][(lane >> 2) * TILE + e];  // head = lane>>2
        float* w = wv + t * (HEADS * HDIM) + lane * 4;
        atomicAdd(w + 0, vv.x * sc);
        atomicAdd(w + 1, vv.y * sc);
        atomicAdd(w + 2, vv.z * sc);
        atomicAdd(w + 3, vv.w * sc);
    }
}

// ---------------------------------------------------------------------------
// Kernel 2: out = wV / (Z + 1e-6), in place on d_out.
// ---------------------------------------------------------------------------
__global__ void ca_norm_kernel(float* __restrict__ wv,
                               const float* __restrict__ z, int n_out) {
    int i = blockIdx.x * blockDim.x + threadIdx.x;
    if (i >= n_out) return;
    const int node = i >> 7;          // / 128
    const int h    = (i >> 4) & 7;    // head
    wv[i] = wv[i] / (z[node * HEADS + h] + 1e-6f);
}

// ---------------------------------------------------------------------------
extern "C" void kernel_launch(void* const* d_in, const int* in_sizes, int n_in,
                              void* d_out, int out_size, void* d_ws, size_t ws_size,
                              hipStream_t stream) {
    const float*   q  = (const float*)d_in[0];
    const float*   k  = (const float*)d_in[1];
    const float*   v  = (const float*)d_in[2];
    const int64_t* ei = (const int64_t*)d_in[3];  // int64 per reference
    float* out = (float*)d_out;                   // [total, 128] = wV accumulator
    float* z   = (float*)d_ws;                    // [total, 8]   = Z accumulator

    const int E     = in_sizes[3] / 2;            // edge_index is [2, E]
    const int total = in_sizes[0] / (HEADS * HDIM);
    const int n_out = total * HEADS * HDIM;
    const int n_z   = total * HEADS;

    const int n_init = (n_out > n_z) ? n_out : n_z;
    ca_zero_kernel<<<(n_init + 255) / 256, 256, 0, stream>>>(out, n_out, z, n_z);

    const int tiles  = (E + TILE - 1) / TILE;
    const int blocks = (tiles + WAVES_PER_BLOCK - 1) / WAVES_PER_BLOCK;
    ca_edge_kernel<<<blocks, WAVES_PER_BLOCK * 32, 0, stream>>>(q, k, v, ei, out, z, E);

    ca_norm_kernel<<<(n_out + 255) / 256, 256, 0, stream>>>(out, z, n_out);
}